// GlottalFlowTable_6949257085235
// MI455X (gfx1250) — compile-verified
//
#include <hip/hip_runtime.h>

#ifndef __has_builtin
#define __has_builtin(x) 0
#endif

#define HOP  512
#define TABN 100

#if __has_builtin(__builtin_amdgcn_global_load_async_to_lds_b128)
// Builtin signature (from hipcc diagnostic): params are pointers to
// 16-byte int vectors in addrspace(1) (global) / addrspace(3) (LDS).
typedef int v4i_t __attribute__((vector_size(16)));
typedef __attribute__((address_space(1))) v4i_t* gptr_v4i;
typedef __attribute__((address_space(3))) v4i_t* lptr_v4i;
#endif

// One workgroup per (batch, frame): blockIdx.x = frame, blockIdx.y = batch
// (2D grid avoids any integer division in the prologue). Stages the two
// 100-float table rows (contiguous 800 B in global memory) into LDS via
// gfx1250 async global->LDS b128 transfers (ASYNCcnt-tracked), re-packs
// them as interleaved float2 pairs so the per-element gathers become
// single 2-address ds_load_b64 ops, then streams the 512 phase values
// with float4 loads/stores. Pure HBM-bandwidth problem (~141 MB total,
// ~6 us at 23.3 TB/s); no matmul structure, so WMMA is inapplicable.
__global__ __launch_bounds__(128)
void glottal_flow_kernel(const float* __restrict__ wp,
                         const float* __restrict__ tables,
                         float* __restrict__ out,
                         int S, int FrP1)
{
    __shared__ float  stg[2 * TABN];   // raw rows f and f+1
    __shared__ float2 tabP[TABN + 4];  // interleaved (row_f[i], row_f1[i]), +wrap

    const int f = blockIdx.x;
    const int b = blockIdx.y;
    const int t = threadIdx.x;

    // rows f and f+1 are contiguous: 200 floats = 800 bytes, 16B-aligned
    // (row start byte offset is a multiple of 400, and 400 % 16 == 0).
    const float* rows = tables + ((size_t)b * (size_t)FrP1 + (size_t)f) * TABN;

#if __has_builtin(__builtin_amdgcn_global_load_async_to_lds_b128)
    // 50 lanes x 16 B = 800 B async DMA into LDS (ASYNCcnt-tracked).
    if (t < 50) {
        __builtin_amdgcn_global_load_async_to_lds_b128(
            (gptr_v4i)(rows + 4 * t),
            (lptr_v4i)&stg[4 * t],
            /*offset=*/0, /*cpol=*/0);
    }
  #if __has_builtin(__builtin_amdgcn_s_wait_asynccnt)
    __builtin_amdgcn_s_wait_asynccnt(0);
  #else
    asm volatile("s_wait_asynccnt 0" ::: "memory");
  #endif
#else
    for (int i = t; i < 2 * TABN; i += 128) stg[i] = rows[i];
#endif
    __syncthreads();

    // Build interleaved pairs; entry TABN is the wraparound (index 100 -> 0).
    if (t <= TABN) {
        const int i = (t == TABN) ? 0 : t;
        tabP[t] = make_float2(stg[i], stg[TABN + i]);
    }
    __syncthreads();

    const size_t base = (size_t)b * (size_t)S + (size_t)f * HOP;
    const float4 x = ((const float4*)(wp + base))[t];

    // Warm L2/L0 for the next frame's phase block (gfx1250 global_prefetch_b8).
    __builtin_prefetch(wp + base + HOP + 4 * t, 0, 1);

    const float xs[4] = {x.x, x.y, x.z, x.w};
    float rs[4];
#pragma unroll
    for (int k = 0; k < 4; ++k) {
        const float raw = xs[k] * (float)TABN;
        int fl = (int)raw;                                   // trunc == astype(int32)
        fl = fl < 0 ? 0 : (fl > TABN - 1 ? TABN - 1 : fl);   // clip(0, 99)
        const float p = raw - (float)fl;

        const float2 c0 = tabP[fl];       // (row_f[fl],   row_f1[fl])
        const float2 c1 = tabP[fl + 1];   // (row_f[fl+1], row_f1[fl+1])

        const float vf = c0.x + p * (c1.x - c0.x);   // table lerp, frame f
        const float vc = c0.y + p * (c1.y - c0.y);   // table lerp, frame f+1
        const float p2 = (float)(4 * t + k) * (1.0f / (float)HOP);
        rs[k] = vf + p2 * (vc - vf);                 // frame lerp
    }
    ((float4*)(out + base))[t] = make_float4(rs[0], rs[1], rs[2], rs[3]);
}

extern "C" void kernel_launch(void* const* d_in, const int* in_sizes, int n_in,
                              void* d_out, int out_size, void* d_ws, size_t ws_size,
                              hipStream_t stream)
{
    const float* wp     = (const float*)d_in[0];
    const float* tables = (const float*)d_in[1];
    float* out = (float*)d_out;

    // Recover shapes: in_sizes[0] = B*S, in_sizes[1] = B*(S/HOP + 1)*TABN.
    const long n0  = (long)in_sizes[0];
    const long n1  = (long)in_sizes[1];
    const long BFr = n0 / HOP;                    // B * Fr (S divisible by HOP here)
    long B = (n1 - BFr * TABN) / TABN;
    if (B <= 0) B = 32;
    const long S  = n0 / B;
    const long Fr = S / HOP;

    dim3 grid((unsigned)Fr, (unsigned)B);
    dim3 block(128);
    hipLaunchKernelGGL(glottal_flow_kernel, grid, block, 0, stream,
                       wp, tables, out, (int)S, (int)(Fr + 1));
}